// Processor_80015240724846
// MI455X (gfx1250) — compile-verified
//
#include <hip/hip_runtime.h>

// GraphConv x5 for MI455X (gfx1250, wave32).
//   layer: agg = scatter_add(h[src]*w -> dst);  h' = agg@W_rel + b_rel + h@W_root; ReLU (not last)
// N = 100000 nodes, E = 1600000 edges, D = 64, 5 layers.
//
// Roofline: scatter dominates (E*D = 102.4M gathered floats + atomics per
// layer; h/agg are 25.6 MB each -> L2-resident on the 192 MB L2). GEMMs are
// ~1.6 GFLOP/layer -> exact-f32 WMMA (V_WMMA_F32_16X16X4_F32) is the right
// precision (no roofline benefit from f16/bf16 here).

#define D 64
#define EDGES_PER_BLOCK 64

typedef __attribute__((ext_vector_type(2))) float v2f;
typedef __attribute__((ext_vector_type(8))) float v8f;

// ---------------------------------------------------------------------------
// Zero the aggregation buffer (float4 stores).
// ---------------------------------------------------------------------------
__global__ void zero_f4_kernel(float4* __restrict__ p, int n4) {
  int i = blockIdx.x * blockDim.x + threadIdx.x;
  if (i < n4) p[i] = make_float4(0.f, 0.f, 0.f, 0.f);
}

// ---------------------------------------------------------------------------
// Edge-parallel scatter, LDS-staged edge metadata:
//   * 64 edges / 256-thread block; threads 0..63 coalesce-load (src,dst,w)
//     and park them in LDS (int32-narrowed ids), removing the 16x redundant
//     per-lane edge loads.
//   * each 16-lane group then processes 4 edges: float4 gather of h[src]
//     (one global_load_b128, coalesced 256B/edge), scale by w, and four
//     global_atomic_add_f32 into agg[dst] (L2-resident).
// ---------------------------------------------------------------------------
__global__ void __launch_bounds__(256)
scatter_kernel(const float* __restrict__ h,
               const long long* __restrict__ src,
               const long long* __restrict__ dst,
               const float* __restrict__ ew,
               float* __restrict__ agg, int n_edges) {
  __shared__ int   sSrc[EDGES_PER_BLOCK];
  __shared__ int   sDst[EDGES_PER_BLOCK];
  __shared__ float sW[EDGES_PER_BLOCK];

  int tid  = threadIdx.x;
  int base = blockIdx.x * EDGES_PER_BLOCK;

  if (tid < EDGES_PER_BLOCK) {
    int e = base + tid;
    int s = 0, d = 0;
    float w = 0.f;                 // w==0 makes OOB edges no-ops
    if (e < n_edges) {
      s = (int)src[e];
      d = (int)dst[e];
      w = ew[e];
    }
    sSrc[tid] = s;
    sDst[tid] = d;
    sW[tid]   = w;
  }
  __syncthreads();

  int lane = tid & 15;             // element group within D (float4)
  int grp  = tid >> 4;             // 0..15: edge group
#pragma unroll
  for (int j = 0; j < 4; ++j) {
    int le = grp + j * 16;         // local edge 0..63
    int s  = sSrc[le];             // LDS broadcast reads
    int d  = sDst[le];
    float w = sW[le];
    float4 v = ((const float4*)(h + (size_t)s * D))[lane];   // global_load_b128
    float* ap = agg + (size_t)d * D + lane * 4;
    atomicAdd(ap + 0, v.x * w);
    atomicAdd(ap + 1, v.y * w);
    atomicAdd(ap + 2, v.z * w);
    atomicAdd(ap + 3, v.w * w);
  }
}

// ---------------------------------------------------------------------------
// Fused dual-GEMM + bias + optional ReLU using FP32 WMMA (exact f32 math):
//   out = agg @ W_rel + b_rel + x @ W_root
// Block = 256 threads = 8 waves; block tile = 32 rows x 64 cols;
// each wave owns one 16x16 output tile; K=64 -> 16 x V_WMMA_F32_16X16X4_F32
// per operand matrix (32 WMMAs per tile). Weights staged in LDS (32 KB of
// the 320 KB WGP LDS). EXEC stays all-1s (no divergent control flow).
//
// VGPR layouts per CDNA5 ISA (7.12.2):
//   A 16x4 f32 : lanes 0-15 -> K=0,1 in a[0],a[1]; lanes 16-31 -> K=2,3.
//   C/D 16x16  : VGPR r: lanes 0-15 = row r, lanes 16-31 = row r+8.
//   B 4x16     : mirrored A layout (lane half selects K pair, lane%16 = col).
// ---------------------------------------------------------------------------
__global__ void __launch_bounds__(256)
gemm_kernel(const float* __restrict__ agg, const float* __restrict__ xin,
            const float* __restrict__ Wrel, const float* __restrict__ brel,
            const float* __restrict__ Wroot, float* __restrict__ out,
            int relu) {
  __shared__ float sWrel[D * D];
  __shared__ float sWroot[D * D];

  int tid = threadIdx.x;
  {
    const float4* g0 = (const float4*)Wrel;
    const float4* g1 = (const float4*)Wroot;
    float4* l0 = (float4*)sWrel;
    float4* l1 = (float4*)sWroot;
#pragma unroll
    for (int j = 0; j < 4; ++j) {
      l0[tid + j * 256] = g0[tid + j * 256];
      l1[tid + j * 256] = g1[tid + j * 256];
    }
  }
  __syncthreads();

  int wave    = tid >> 5;    // 0..7
  int lane    = tid & 31;
  int rlane   = lane & 15;
  int khalf   = lane >> 4;   // 0 or 1 -> which K pair of the 4-wide step
  int rowTile = wave >> 2;   // 0..1
  int colTile = wave & 3;    // 0..3
  int rowBase = blockIdx.x * 32 + rowTile * 16;
  int col     = colTile * 16 + rlane;

  // Accumulator pre-loaded with bias (broadcast along rows: C[m][col]=b[col]).
  float bias = brel[col];
  v8f c;
#pragma unroll
  for (int r = 0; r < 8; ++r) c[r] = bias;

  const float* Arow = agg + (size_t)(rowBase + rlane) * D;
  const float* Xrow = xin + (size_t)(rowBase + rlane) * D;

#pragma unroll
  for (int s = 0; s < 16; ++s) {
    int k = 4 * s + 2 * khalf;
    v2f a = *(const v2f*)(Arow + k);                 // global_load_b64
    v2f b;
    b[0] = sWrel[k * D + col];                       // ds_load_2addr_stride64
    b[1] = sWrel[(k + 1) * D + col];
    c = __builtin_amdgcn_wmma_f32_16x16x4_f32(
        /*neg_a=*/false, a, /*neg_b=*/false, b,
        /*c_mod=*/(short)0, c, /*reuse_a=*/false, /*reuse_b=*/false);
  }
#pragma unroll
  for (int s = 0; s < 16; ++s) {
    int k = 4 * s + 2 * khalf;
    v2f a = *(const v2f*)(Xrow + k);
    v2f b;
    b[0] = sWroot[k * D + col];
    b[1] = sWroot[(k + 1) * D + col];
    c = __builtin_amdgcn_wmma_f32_16x16x4_f32(
        false, a, false, b, (short)0, c, false, false);
  }

  int mBase = rowBase + 8 * khalf;
#pragma unroll
  for (int r = 0; r < 8; ++r) {
    float v = c[r];
    if (relu) v = fmaxf(v, 0.f);   // uniform SGPR branch, EXEC unchanged
    out[(size_t)(mBase + r) * D + col] = v;
  }
}

// ---------------------------------------------------------------------------
// Host-side launch sequence (graph-capture safe: only kernel launches).
// Inputs: x[N,64] f32, edge_index[2,E] int64, edge_weight[E] f32,
//         W_rel[5,64,64] f32, b_rel[5,64] f32, W_root[5,64,64] f32.
// ---------------------------------------------------------------------------
extern "C" void kernel_launch(void* const* d_in, const int* in_sizes, int n_in,
                              void* d_out, int out_size, void* d_ws, size_t ws_size,
                              hipStream_t stream) {
  (void)n_in; (void)out_size; (void)ws_size;

  const float*     x     = (const float*)d_in[0];
  const long long* ei    = (const long long*)d_in[1];  // int64 per reference
  const float*     ew    = (const float*)d_in[2];
  const float*     Wrel  = (const float*)d_in[3];
  const float*     brel  = (const float*)d_in[4];
  const float*     Wroot = (const float*)d_in[5];
  float*           out   = (float*)d_out;

  const int N = in_sizes[0] / D;   // 100000 (divisible by 32)
  const int E = in_sizes[2];       // 1600000 (divisible by 64)
  const long long* src = ei;       // edge_index[0,:]
  const long long* dst = ei + E;   // edge_index[1,:]

  float* ha  = (float*)d_ws;                // h ping
  float* hb  = ha + (size_t)N * D;          // h pong
  float* agg = hb + (size_t)N * D;          // aggregation buffer

  const int n4       = N * (D / 4);
  const int zeroGrid = (n4 + 255) / 256;
  const int scatGrid = (E + EDGES_PER_BLOCK - 1) / EDGES_PER_BLOCK;
  const int gemmGrid = N / 32;              // 32 rows / block, exact

  const float* cur = x;
  for (int layer = 0; layer < 5; ++layer) {
    zero_f4_kernel<<<zeroGrid, 256, 0, stream>>>((float4*)agg, n4);
    scatter_kernel<<<scatGrid, 256, 0, stream>>>(cur, src, dst, ew, agg, E);
    float* o = (layer == 4) ? out : ((layer & 1) ? hb : ha);
    gemm_kernel<<<gemmGrid, 256, 0, stream>>>(
        agg, cur, Wrel + (size_t)layer * D * D, brel + (size_t)layer * D,
        Wroot + (size_t)layer * D * D, o, (layer < 4) ? 1 : 0);
    cur = o;
  }
}